// FineMatchingLoss_66666482369254
// MI455X (gfx1250) — compile-verified
//
#include <hip/hip_runtime.h>
#include <hip/hip_bf16.h>

typedef __attribute__((ext_vector_type(2))) float v2f;
typedef __attribute__((ext_vector_type(8))) float v8f;

#define KNN 64
#define RAD2 0.01f          // POSITIVE_RADIUS^2
#define WAVES_PER_BLOCK 8

// One wave32 per correspondence c.
__global__ __launch_bounds__(256) void fml_main(
    const float* __restrict__ ref_pts,          // C*64*3 f32
    const float* __restrict__ src_pts,          // C*64*3 f32
    const unsigned char* __restrict__ ref_masks,// C*64 bool(1B)
    const unsigned char* __restrict__ src_masks,// C*64 bool(1B)
    const float* __restrict__ scores,           // C*65*65 f32
    const float* __restrict__ transform,        // B*16 f32
    const int* __restrict__ src_idx,            // C int32
    const long long* __restrict__ src_len,      // B int64
    float* __restrict__ ws,                     // 2*B accumulators (sum,cnt)
    int C, int Bn)
{
    // Packed B-operand staging: s_bx[0..63] = -2*sx, s_bx[64..127] = -2*sz
    __shared__ float s_bx[WAVES_PER_BLOCK][2 * KNN];
    __shared__ float s_by[WAVES_PER_BLOCK][KNN];      // -2*sy
    __shared__ float s_sq[WAVES_PER_BLOCK][KNN];      // |s|^2

    const int wave = (int)((blockIdx.x * blockDim.x + threadIdx.x) >> 5);
    const int lane = (int)(threadIdx.x & 31);
    const int wl   = (int)(threadIdx.x >> 5);
    if (wave >= C) return;  // wave-uniform exit; no __syncthreads used anywhere

    // ---- segment id: seg = count(boundaries <= idx), boundaries = cumsum(src_len)
    const int idx = src_idx[wave];
    int seg = 0;
    long long cum = 0;
    for (int b = 0; b < Bn; ++b) {
        cum += src_len[b];
        seg += (idx >= cum) ? 1 : 0;
    }
    if (seg >= Bn) seg = Bn - 1;

    // ---- masks as wave-uniform bitmasks (4 byte-loads + 4 ballots per wave)
    const unsigned char* rm = ref_masks + (size_t)wave * KNN;
    const unsigned char* sm = src_masks + (size_t)wave * KNN;
    const unsigned rmask0 = (unsigned)__ballot(rm[lane]      != 0); // rows 0..31
    const unsigned rmask1 = (unsigned)__ballot(rm[lane + 32] != 0); // rows 32..63
    const unsigned smask0 = (unsigned)__ballot(sm[lane]      != 0); // cols 0..31
    const unsigned smask1 = (unsigned)__ballot(sm[lane + 32] != 0); // cols 32..63

    // ---- per-segment transform (broadcast load, cached)
    const float* T = transform + (size_t)seg * 16;
    const float R00 = T[0],  R01 = T[1],  R02 = T[2],  t0 = T[3];
    const float R10 = T[4],  R11 = T[5],  R12 = T[6],  t1 = T[7];
    const float R20 = T[8],  R21 = T[9],  R22 = T[10], t2 = T[11];

    // ---- phase 1: transform src points, pre-scale by -2, stage in LDS
    const float* sp = src_pts + (size_t)wave * KNN * 3;
    #pragma unroll
    for (int r = 0; r < 2; ++r) {
        const int k = lane + r * 32;
        const float x = sp[k * 3 + 0];
        const float y = sp[k * 3 + 1];
        const float z = sp[k * 3 + 2];
        const float tx = R00 * x + R01 * y + R02 * z + t0;
        const float ty = R10 * x + R11 * y + R12 * z + t1;
        const float tz = R20 * x + R21 * y + R22 * z + t2;
        s_bx[wl][k]       = -2.0f * tx;
        s_bx[wl][k + KNN] = -2.0f * tz;
        s_by[wl][k]       = -2.0f * ty;
        s_sq[wl][k]       = tx * tx + ty * ty + tz * tz;
    }

    const int half = lane >> 4;   // 0: K=0,1 ; 1: K=2,3 (A/B frag halves)
    const int l15  = lane & 15;   // row (A) / col (B,C,D) within tile

    // ---- preload all 4 B fragments + C values into registers (same-wave
    //      LDS ordering guarantees the phase-1 stores are visible)
    v2f   bfr[4];
    float csr[4];
    unsigned smsel[4];            // 0xFF if src_mask[j] else 0
    #pragma unroll
    for (int tj = 0; tj < 4; ++tj) {
        const int j = tj * 16 + l15;
        bfr[tj].x = s_bx[wl][half * KNN + j];               // -2sx or -2sz
        bfr[tj].y = (half == 0) ? s_by[wl][j] : 1.0f;       // -2sy or 1
        csr[tj]   = s_sq[wl][j];
        const unsigned ssel = (tj < 2) ? smask0 : smask1;
        smsel[tj] = (((ssel >> (((tj & 1) << 4) + l15)) & 1u) != 0) ? 0xFFu : 0u;
    }

    // ---- phase 2: 4x4 tiles of 16x16 squared distances via V_WMMA_F32_16X16X4_F32
    // A row_i = [rx, ry, rz, |r|^2], B col_j = [-2sx, -2sy, -2sz, 1], C = |s_j|^2
    const float* rp = ref_pts + (size_t)wave * KNN * 3;
    const float* sc = scores  + (size_t)wave * 65 * 65;

    float lsum = 0.0f;
    int   icnt = 0;
    unsigned long long rowacc = 0ull;   // per-lane row-hit bits (actual row index)
    unsigned long long colacc = 0ull;   // per-lane col-hit bits (actual col index)

    #pragma unroll
    for (int ti = 0; ti < 4; ++ti) {
        const int i0 = ti * 16 + l15;
        const float rx = rp[i0 * 3 + 0];
        const float ry = rp[i0 * 3 + 1];
        const float rz = rp[i0 * 3 + 2];
        v2f a;
        a.x = (half == 0) ? rx : rz;
        a.y = (half == 0) ? ry : (rx * rx + ry * ry + rz * rz);

        const int ibase = ti * 16 + (half << 3);   // first D row this lane holds
        // 8-bit slice of ref-mask covering rows ibase + [0..7]
        const unsigned rsel  = (ti < 2) ? rmask0 : rmask1;
        const unsigned rbits = (rsel >> (ibase & 31)) & 0xFFu;

        #pragma unroll
        for (int tj = 0; tj < 4; ++tj) {
            const int j = tj * 16 + l15;

            // D = A*B + C  -> squared distances for this 16x16 tile
            v8f cf = { csr[tj], csr[tj], csr[tj], csr[tj],
                       csr[tj], csr[tj], csr[tj], csr[tj] };
            v8f d = __builtin_amdgcn_wmma_f32_16x16x4_f32(
                false, a, false, bfr[tj], (short)0, cf, false, false);

            // branch-free pack of 8 threshold tests into a hit mask
            unsigned hitbits = 0u;
            #pragma unroll
            for (int v = 0; v < 8; ++v)
                hitbits |= (d[v] < RAD2) ? (1u << v) : 0u;
            hitbits &= rbits & smsel[tj];

            rowacc |= ((unsigned long long)hitbits) << ibase;
            colacc |= hitbits ? (1ull << j) : 0ull;

            if (hitbits) {   // rare (~0.4% of pairs)
                icnt += __popc(hitbits);
                unsigned bits = hitbits;
                while (bits) {
                    const int v = __ffs(bits) - 1;
                    bits &= bits - 1u;
                    lsum -= sc[(ibase + v) * 65 + j];
                }
            }
        }
    }

    // ---- OR-reduce row/col hit masks across the wave (becomes uniform)
    #pragma unroll
    for (int off = 16; off >= 1; off >>= 1) {
        rowacc |= (unsigned long long)__shfl_xor((unsigned long long)rowacc, off, 32);
        colacc |= (unsigned long long)__shfl_xor((unsigned long long)colacc, off, 32);
    }

    // ---- phase 3: slack row/col entries — pure register bit tests
    #pragma unroll
    for (int r = 0; r < 2; ++r) {
        const int k = lane + r * 32;
        const unsigned rmk = (((r == 0) ? rmask0 : rmask1) >> lane) & 1u;
        const unsigned smk = (((r == 0) ? smask0 : smask1) >> lane) & 1u;
        if ((((rowacc >> k) & 1ull) == 0ull) && rmk) {
            lsum -= sc[k * 65 + 64];
            icnt += 1;
        }
        if ((((colacc >> k) & 1ull) == 0ull) && smk) {
            lsum -= sc[64 * 65 + k];
            icnt += 1;
        }
    }

    // ---- phase 4: wave32 reduction, then 2 global float atomics per wave
    float lcnt = (float)icnt;
    #pragma unroll
    for (int off = 16; off >= 1; off >>= 1) {
        lsum += __shfl_xor(lsum, off, 32);
        lcnt += __shfl_xor(lcnt, off, 32);
    }
    if (lane == 0) {
        atomicAdd(&ws[seg * 2 + 0], lsum);
        atomicAdd(&ws[seg * 2 + 1], lcnt);
    }
}

__global__ void fml_init(float* __restrict__ ws, int n) {
    const int i = (int)(blockIdx.x * blockDim.x + threadIdx.x);
    if (i < n) ws[i] = 0.0f;
}

__global__ void fml_final(const float* __restrict__ ws, float* __restrict__ out, int Bn) {
    if (threadIdx.x == 0 && blockIdx.x == 0) {
        float acc = 0.0f;
        float n = 0.0f;
        for (int b = 0; b < Bn; ++b) {
            const float s = ws[2 * b + 0];
            const float c = ws[2 * b + 1];
            if (c > 0.0f) { acc += s / c; n += 1.0f; }
        }
        out[0] = (n > 0.0f) ? (acc / n) : 0.0f;
    }
}

extern "C" void kernel_launch(void* const* d_in, const int* in_sizes, int n_in,
                              void* d_out, int out_size, void* d_ws, size_t ws_size,
                              hipStream_t stream) {
    const float*         ref_pts   = (const float*)d_in[0];
    const float*         src_pts   = (const float*)d_in[1];
    const unsigned char* ref_masks = (const unsigned char*)d_in[2];
    const unsigned char* src_masks = (const unsigned char*)d_in[3];
    const float*         scores    = (const float*)d_in[4];
    const float*         transform = (const float*)d_in[5];
    const int*           src_idx   = (const int*)d_in[6];
    const long long*     src_len   = (const long long*)d_in[7];

    const int C  = in_sizes[4] / (65 * 65);
    const int Bn = in_sizes[5] / 16;
    float* ws  = (float*)d_ws;
    float* out = (float*)d_out;

    fml_init<<<1, 64, 0, stream>>>(ws, 2 * Bn);

    const int blocks = (C + WAVES_PER_BLOCK - 1) / WAVES_PER_BLOCK;
    fml_main<<<blocks, 32 * WAVES_PER_BLOCK, 0, stream>>>(
        ref_pts, src_pts, ref_masks, src_masks, scores, transform,
        src_idx, src_len, ws, C, Bn);

    fml_final<<<1, 1, 0, stream>>>(ws, out, Bn);
}